// DCT2d_10728828305641
// MI455X (gfx1250) — compile-verified
//
#include <hip/hip_runtime.h>

typedef __attribute__((ext_vector_type(2))) float v2f;
typedef __attribute__((ext_vector_type(4))) float v4f;
typedef __attribute__((ext_vector_type(8))) float v8f;

#define IMG_W 1024
#define IMG_H 1024
#define NTILES_X 64              // 1024 / 16
#define TILES_PER_IMG (64 * 64)  // 4096
#define WAVES_PER_BLOCK 8

// One wave32 per 16x16 tile (= 2x2 grid of 8x8 DCT blocks).
// D = Mb * T * Mb^T with Mb = diag(A, A) computes all four 8x8 DCTs at once.
__global__ __launch_bounds__(256) void dct8x8_wmma_kernel(
    const float* __restrict__ x,   // (32, 1, 1024, 1024) f32
    const float* __restrict__ A,   // (8, 8) f32 DCT matrix
    float* __restrict__ out)       // (32, 16384, 8, 8) f32
{
    // Per-wave LDS scratch: W matrix (row stride 20 to kill bank conflicts
    // between half-wave address sets), and D staged in output order.
    __shared__ float ldsW[WAVES_PER_BLOCK][16 * 20];
    __shared__ float ldsD[WAVES_PER_BLOCK][256];

    const int lane = threadIdx.x & 31;
    const int wv   = threadIdx.x >> 5;     // wave id within block
    const int row  = lane & 15;            // M (or N) index 0..15
    const int kh   = lane >> 4;            // half-wave selector (K split)

    // Global tile id -> (image n, tile y, tile x)
    const int tile = blockIdx.x * WAVES_PER_BLOCK + wv;
    const int n    = tile / TILES_PER_IMG;
    const int t    = tile - n * TILES_PER_IMG;
    const int ty   = t >> 6;
    const int tx   = t & 63;

    const float* xt = x + (long long)n * (IMG_W * IMG_H)
                        + (long long)(ty * 16) * IMG_W + tx * 16;

    // ---- Mb fragments --------------------------------------------------
    // A-layout slice kc (16x4): lane holds Mb[row, kc*4 + kh*2 + v], v=0,1.
    // By layout symmetry this is ALSO the B-layout fragment of Mb^T.
    // Mb[r,k] = (r/8 == k/8) ? A[r%8, k%8] : 0
    v2f mfrag[4];
#pragma unroll
    for (int kc = 0; kc < 4; ++kc) {
#pragma unroll
        for (int v = 0; v < 2; ++v) {
            const int k = kc * 4 + kh * 2 + v;
            mfrag[kc][v] = ((row >> 3) == (k >> 3)) ? A[(row & 7) * 8 + (k & 7)]
                                                    : 0.0f;
        }
    }

    // ---- W = T * Mb^T  (4 x v_wmma_f32_16x16x4_f32) --------------------
    // T in A-layout: lane loads T[row, kc*4 + kh*2 + {0,1}] -> one b64 load.
    v8f wacc = {};
#pragma unroll
    for (int kc = 0; kc < 4; ++kc) {
        const v2f tfrag =
            *(const v2f*)(xt + (long long)row * IMG_W + kc * 4 + kh * 2);
        wacc = __builtin_amdgcn_wmma_f32_16x16x4_f32(
            false, tfrag, false, mfrag[kc], (short)0, wacc, false, false);
    }

    // ---- layout turn through LDS: C-layout -> B-layout ------------------
    // C-layout: vgpr r holds W[r + 8*kh][row].
    float* lw = ldsW[wv];
#pragma unroll
    for (int r = 0; r < 8; ++r)
        lw[(r + 8 * kh) * 20 + row] = wacc[r];
    // Same-wave LDS ops are in-order; compiler inserts s_wait_dscnt before use.

    // ---- D = Mb * W  (4 x v_wmma_f32_16x16x4_f32) -----------------------
    v8f dacc = {};
#pragma unroll
    for (int kc = 0; kc < 4; ++kc) {
        const int kbase = kc * 4 + kh * 2;
        v2f wfrag;
        wfrag[0] = lw[(kbase + 0) * 20 + row];
        wfrag[1] = lw[(kbase + 1) * 20 + row];
        dacc = __builtin_amdgcn_wmma_f32_16x16x4_f32(
            false, mfrag[kc], false, wfrag, (short)0, dacc, false, false);
    }

    // ---- stage D in output (block-major) order --------------------------
    // D[i_g = r + 8*kh][m_g = row]; within-tile output offset:
    //   bi*128 + bj*64 + i*8 + m, bi = i_g/8 = kh, bj = m_g/8, i = i_g%8 = r.
    float* ld = ldsD[wv];
    const int obase = kh * 128 + (row >> 3) * 64 + (row & 7);
#pragma unroll
    for (int r = 0; r < 8; ++r)
        ld[obase + r * 8] = dacc[r];

    // ---- coalesced b128 stores: two contiguous 512B chunks per tile -----
    // Chunk bi holds blocks (ty*2+bi, tx*2) and (ty*2+bi, tx*2+1): 128 floats.
#pragma unroll
    for (int bi = 0; bi < 2; ++bi) {
        const long long go = (long long)n * 1048576
                           + (long long)((ty * 2 + bi) * 128 + tx * 2) * 64
                           + lane * 4;
        *(v4f*)(out + go) = *(const v4f*)(ld + bi * 128 + lane * 4);
    }
}

extern "C" void kernel_launch(void* const* d_in, const int* in_sizes, int n_in,
                              void* d_out, int out_size, void* d_ws, size_t ws_size,
                              hipStream_t stream) {
    const float* x = (const float*)d_in[0];   // (32,1,1024,1024) f32
    const float* A = (const float*)d_in[1];   // (8,8) f32
    float* out = (float*)d_out;               // (32,16384,8,8) f32

    const int n_imgs = in_sizes[0] / (IMG_W * IMG_H);     // 32
    const int total_tiles = n_imgs * TILES_PER_IMG;       // 131072
    const int blocks = total_tiles / WAVES_PER_BLOCK;     // 16384

    dct8x8_wmma_kernel<<<blocks, 256, 0, stream>>>(x, A, out);
}